// BagAttention_27092653703393
// MI455X (gfx1250) — compile-verified
//
#include <hip/hip_runtime.h>
#include <hip/hip_bf16.h>
#include <math.h>

#define NUM_BAGS 16384
#define IN_DIM   768
#define MAX_BAG  16
#define THREADS  192              // IN_DIM / 4: one float4 column slice per thread
#define NWAVES   (THREADS / 32)   // 6 wave32s

#if defined(__gfx1250__) && __has_builtin(__builtin_amdgcn_global_load_async_to_lds_b128)
#define HAVE_ASYNC_LDS 1
typedef int v4i __attribute__((vector_size(16)));           // matches builtin's int4 pointee
typedef __attribute__((address_space(1))) v4i* gvec_p;      // __device__ (global) int4*
typedef __attribute__((address_space(3))) v4i* lvec_p;      // __shared__ (LDS) int4*
#endif

#if defined(__gfx1250__) && __has_builtin(__builtin_amdgcn_wmma_f32_16x16x4_f32)
#define HAVE_WMMA_F32 1
typedef float v2f __attribute__((ext_vector_type(2)));
typedef float v8f __attribute__((ext_vector_type(8)));
#endif

__global__ __launch_bounds__(THREADS) void bag_attention_kernel(
    const float* __restrict__ x,        // [TOTAL, 768]
    const float* __restrict__ attn_w,   // [768]
    const int*   __restrict__ scope,    // [NUM_BAGS] cumulative end offsets
    float* __restrict__ out_logits,     // [NUM_BAGS, 768]
    float* __restrict__ out_attn)       // [TOTAL]
{
    __shared__ float lds_x[MAX_BAG * IN_DIM];        // 48 KB: whole bag
    __shared__ float lds_part[MAX_BAG][NWAVES];      // per-wave dot partials
    __shared__ float lds_attn[MAX_BAG];              // softmax weights (zero-padded to 16)

    const int bag   = blockIdx.x;
    const int t     = threadIdx.x;
    const int start = (bag == 0) ? 0 : scope[bag - 1];
    const int size  = scope[bag] - start;            // 1..16

    // ---- Phase A: bag rows -> LDS (async global->LDS on gfx1250) ----
    const float* src = x + (size_t)start * IN_DIM + t * 4;
    float*       dst = lds_x + t * 4;
    for (int r = 0; r < size; ++r) {
#ifdef HAVE_ASYNC_LDS
        __builtin_amdgcn_global_load_async_to_lds_b128(
            (gvec_p)(src + r * IN_DIM), (lvec_p)(dst + r * IN_DIM), 0, 0);
#else
        *(float4*)(dst + r * IN_DIM) = *(const float4*)(src + r * IN_DIM);
#endif
    }
#ifdef HAVE_ASYNC_LDS
#if __has_builtin(__builtin_amdgcn_s_wait_asynccnt)
    __builtin_amdgcn_s_wait_asynccnt(0);
#else
    asm volatile("s_wait_asynccnt 0" ::: "memory");
#endif
#endif
    __syncthreads();

    // ---- Phase B: per-row dot(x_r, w), wave32 shuffle reduction (GEMV, N=1:
    //      WMMA would waste 15/16 lanes here; VALU+shfl is the right tool) ----
    const float4 wv  = *(const float4*)(attn_w + t * 4);
    const int lane   = t & 31;
    const int waveid = t >> 5;
    for (int r = 0; r < size; ++r) {
        const float4 xv = *(const float4*)(lds_x + r * IN_DIM + t * 4);
        float p = xv.x * wv.x + xv.y * wv.y + xv.z * wv.z + xv.w * wv.w;
        #pragma unroll
        for (int off = 16; off > 0; off >>= 1)
            p += __shfl_xor(p, off, 32);
        if (lane == 0) lds_part[r][waveid] = p;
    }
    __syncthreads();

    // ---- thread 0: combine partials, stable softmax over <=16 logits ----
    if (t == 0) {
        const float scale = 0.03608439182435161f;    // 768^-0.5
        float logit[MAX_BAG];
        float m = -INFINITY;
        for (int r = 0; r < size; ++r) {
            float s = 0.0f;
            #pragma unroll
            for (int w2 = 0; w2 < NWAVES; ++w2) s += lds_part[r][w2];
            logit[r] = s * scale;
            m = fmaxf(m, logit[r]);
        }
        float denom = 0.0f;
        for (int r = 0; r < size; ++r) {
            const float e = expf(logit[r] - m);
            lds_attn[r] = e;
            denom += e;
        }
        for (int r = 0; r < size; ++r) lds_attn[r] /= denom;
        for (int r = size; r < MAX_BAG; ++r) lds_attn[r] = 0.0f;  // zero-pad K to 16
    }
    __syncthreads();

    // ---- write per-row softmax weights ----
    if (t < size) out_attn[start + t] = lds_attn[t];

    // ---- Phase C: pooling  out[1x768] = attn[1x16] (x) X[16x768] ----
#ifdef HAVE_WMMA_F32
    // v_wmma_f32_16x16x4_f32, attn replicated across M (all 16 D rows identical).
    // A 16x4 f32 layout (ISA 7.12.2): VGPR0 = K0(lo)/K2(hi), VGPR1 = K1(lo)/K3(hi).
    // B 4x16 mirrors the K=i / K=i+2 half-split; B rows striped across lanes.
    {
        const int half = (lane >> 4) & 1;             // lane half selects K slot
        const int ncol = lane & 15;
        #pragma unroll 2
        for (int j = 0; j < 8; ++j) {                 // 48 n-chunks over 6 waves
            const int nb = waveid * 8 + j;
            const int n  = nb * 16 + ncol;
            v8f c = {};
            #pragma unroll
            for (int kb = 0; kb < 4; ++kb) {          // K = 16 in 4 steps of 4
                const int k0 = kb * 4 + half * 2;
                const int k1 = k0 + 1;
                const int r0 = (k0 < size) ? k0 : 0;  // clamp: pad rows read row 0,
                const int r1 = (k1 < size) ? k1 : 0;  // attn pad = 0 => contributes 0
                v2f a, b;
                a.x = lds_attn[k0];
                a.y = lds_attn[k1];
                b.x = lds_x[r0 * IN_DIM + n];
                b.y = lds_x[r1 * IN_DIM + n];
                c = __builtin_amdgcn_wmma_f32_16x16x4_f32(
                        false, a, false, b, (short)0, c, false, false);
            }
            // D row M=0: VGPR0, lanes 0-15 hold N=0..15
            if (lane < 16)
                out_logits[(size_t)bag * IN_DIM + nb * 16 + lane] = c[0];
        }
    }
#else
    {
        float4 acc = make_float4(0.f, 0.f, 0.f, 0.f);
        for (int r = 0; r < size; ++r) {
            const float  a  = lds_attn[r];
            const float4 xv = *(const float4*)(lds_x + r * IN_DIM + t * 4);
            acc.x += a * xv.x;
            acc.y += a * xv.y;
            acc.z += a * xv.z;
            acc.w += a * xv.w;
        }
        *(float4*)(out_logits + (size_t)bag * IN_DIM + t * 4) = acc;
    }
#endif
}

extern "C" void kernel_launch(void* const* d_in, const int* in_sizes, int n_in,
                              void* d_out, int out_size, void* d_ws, size_t ws_size,
                              hipStream_t stream) {
    const float* x      = (const float*)d_in[0];
    const float* attn_w = (const float*)d_in[1];
    const int*   scope  = (const int*)d_in[2];

    float* out        = (float*)d_out;
    float* out_logits = out;                               // [16384*768]
    float* out_attn   = out + (size_t)NUM_BAGS * IN_DIM;   // [139264]

    bag_attention_kernel<<<NUM_BAGS, THREADS, 0, stream>>>(
        x, attn_w, scope, out_logits, out_attn);
}